// MultiHeadedSpatialAttention_43499428773964
// MI455X (gfx1250) — compile-verified
//
#include <hip/hip_runtime.h>

typedef __attribute__((ext_vector_type(2))) float v2f;
typedef __attribute__((ext_vector_type(8))) float v8f;

#define B_   8
#define C_   64
#define H_   64
#define W_   64
#define HW_  (H_ * W_)
#define TH   8            // output rows per workgroup
#define TROWS (TH + 6)    // 14 rows including 7x7 halo
#define CH_STRIDE 900     // 14*64 = 896 floats + 4 pad (lanes 16-31 land on disjoint LDS banks)

// One workgroup = (batch b, group-pair gp, 8-row band).
// Phase 1: v = blockdiag(Wv[2gp], Wv[2gp+1]) * x  via chained V_WMMA_F32_16X16X4_F32 -> LDS
// Phase 2: separable 7x7 zero-padded box sum (horizontal 7-sums into registers,
//          then 7-row vertical sums) -> out
__global__ __launch_bounds__(256)
void mhsa_fused_wmma(const float* __restrict__ x,
                     const float* __restrict__ Wv,
                     float* __restrict__ out)
{
    __shared__ float vbuf[16 * CH_STRIDE];   // 57.6 KB

    const int bid = blockIdx.x;
    const int ht  = bid & 7;          // 8 row-bands
    const int gp  = (bid >> 3) & 3;   // 4 group pairs
    const int b   = bid >> 5;         // 8 batches
    const int h0  = ht * TH;

    const int lane = threadIdx.x & 31;
    const int wave = threadIdx.x >> 5;     // 8 waves per block
    const int n    = lane & 15;
    const int half = lane >> 4;

    // ---- A operands: block-diagonal [Wv[2gp] 0; 0 Wv[2gp+1]] (16x16), split into 4 K-chunks of 4.
    // ISA layout (16x4 f32 A): lane m = lane&15; VGPRs hold K = kb*4 + 2*(lane>>4) + {0,1}.
    v2f a[4];
    {
        const int m = n;
        #pragma unroll
        for (int kb = 0; kb < 4; ++kb) {
            const int k0 = kb * 4 + half * 2;
            #pragma unroll
            for (int j = 0; j < 2; ++j) {
                const int k = k0 + j;
                float wv = 0.0f;
                if ((m < 8) == (k < 8)) {                 // diagonal 8x8 blocks only
                    const int g = 2 * gp + (m >> 3);
                    wv = Wv[g * 64 + (m & 7) * 8 + (k & 7)];
                }
                a[kb][j] = wv;
            }
        }
    }

    // ---- Phase 1: WMMA-compute v for TROWS x 64 pixels, 16 channels.
    // 56 pixel-tiles of 16 contiguous columns; wave w owns tiles [7w, 7w+7).
    const float* xb = x + ((size_t)b * C_ + gp * 16) * HW_;
    #pragma unroll 1
    for (int tt = wave * 7; tt < wave * 7 + 7; ++tt) {
        const int lr = tt >> 2;              // local row 0..13
        const int gr = h0 - 3 + lr;          // global row (may be OOB -> zero)
        const int wc = (tt & 3) * 16 + n;    // column for this lane (N index)

        v2f bm[4];
        if (gr >= 0 && gr < H_) {            // wave-uniform branch, gates loads only
            const float* xp = xb + gr * W_ + wc;
            #pragma unroll
            for (int kb = 0; kb < 4; ++kb) {
                const int c0 = kb * 4 + half * 2;   // K (input-channel) index, B layout mirrors A
                bm[kb][0] = xp[(size_t)(c0 + 0) * HW_];
                bm[kb][1] = xp[(size_t)(c0 + 1) * HW_];
            }
        } else {
            #pragma unroll
            for (int kb = 0; kb < 4; ++kb) { bm[kb][0] = 0.0f; bm[kb][1] = 0.0f; }
        }

        v8f acc = {};
        #pragma unroll
        for (int kb = 0; kb < 4; ++kb) {     // EXEC is all-ones here: WMMA is unconditional
            acc = __builtin_amdgcn_wmma_f32_16x16x4_f32(
                    false, a[kb], false, bm[kb], (short)0, acc, false, false);
        }

        // D layout: VGPR r, lanes 0-15 -> channel r, lanes 16-31 -> channel r+8; N = lane&15.
        #pragma unroll
        for (int r = 0; r < 8; ++r) {
            const int ch = r + half * 8;
            vbuf[ch * CH_STRIDE + tt * 16 + n] = acc[r];
        }
    }

    __syncthreads();

    // ---- Phase 2: separable 7x7 box sum from LDS.
    // Each thread owns (ch, w) column strips: 16ch * 64w = 1024 strips / 256 threads = 4 each.
    // Per strip: 14 horizontal 7-sums (7 LDS reads each, column-bounded), then 8 vertical
    // 7-sums from registers. 98 LDS reads per 8 outputs vs 392 for the direct 49-tap.
    #pragma unroll 1
    for (int j = threadIdx.x; j < 16 * W_; j += 256) {
        const int ch = j >> 6;
        const int w  = j & 63;

        float hs[TROWS];
        #pragma unroll
        for (int lr = 0; lr < TROWS; ++lr) {
            const float* row = vbuf + ch * CH_STRIDE + lr * W_;
            float s = 0.0f;
            #pragma unroll
            for (int dc = -3; dc <= 3; ++dc) {
                const int w2 = w + dc;
                if (w2 >= 0 && w2 < W_) s += row[w2];
            }
            hs[lr] = s;
        }

        float* op = out + (((size_t)b * C_ + gp * 16 + ch) * H_ + h0) * W_ + w;
        #pragma unroll
        for (int r = 0; r < TH; ++r) {       // buffer rows r..r+6 == global h0+r-3 .. h0+r+3
            float s = hs[r];
            #pragma unroll
            for (int dr = 1; dr < 7; ++dr) s += hs[r + dr];
            op[(size_t)r * W_] = s;          // lane == w: fully coalesced per row
        }
    }
}

extern "C" void kernel_launch(void* const* d_in, const int* in_sizes, int n_in,
                              void* d_out, int out_size, void* d_ws, size_t ws_size,
                              hipStream_t stream) {
    // setup_inputs order: x, Wq, Wk, Wv  (Wq/Wk are dead: softmax over a singleton axis == 1)
    const float* x  = (const float*)d_in[0];
    const float* Wv = (const float*)d_in[3];
    float* out = (float*)d_out;

    dim3 grid(B_ * 4 * (H_ / TH));   // 8 batches * 4 group-pairs * 8 row-bands = 256 blocks
    mhsa_fused_wmma<<<grid, 256, 0, stream>>>(x, Wv, out);

    (void)in_sizes; (void)n_in; (void)d_ws; (void)ws_size; (void)out_size;
}